// GhostMassOptimizer_21869973471543
// MI455X (gfx1250) — compile-verified
//
#include <hip/hip_runtime.h>
#include <math.h>

#define EPSC  1e-4f
#define EPS2  1e-8f        // EPSC^2
#define IE3   1e12f        // 1/EPSC^3
#define LRC   0.05f
#define B1C   0.9f
#define B2C   0.999f
#define AEPS  1e-8f
#define NB    8
#define NN    1024
#define BNT   (NB*NN)      // 8192

typedef __attribute__((ext_vector_type(2))) float v2f;
typedef __attribute__((ext_vector_type(8))) float v8f;

// ---------- small device helpers ----------
__device__ __forceinline__ float sp_f(float x){ return x > 20.f ? x : log1pf(expf(x)); }
__device__ __forceinline__ float sig_f(float x){ return 1.f/(1.f+expf(-x)); }
__device__ __forceinline__ unsigned h32(unsigned x){
  x ^= x>>16; x *= 0x7feb352dU; x ^= x>>15; x *= 0x846ca68bU; x ^= x>>16; return x;
}
__device__ __forceinline__ float u01(unsigned x){ return ((float)(x>>8) + 0.5f)*(1.f/16777216.f); }
__device__ __forceinline__ float nrm_f(unsigned s){
  float u1 = u01(h32(s*2u+1u)), u2 = u01(h32(s*2u+2u));
  return sqrtf(-2.f*logf(u1))*cosf(6.2831853f*u2);
}
// 1/max(d,eps)^3 from r^2 and its rsqrt, branchless (no div/sqrt)
__device__ __forceinline__ float inv3_of(float r2, float rsq){
  return (r2 >= EPS2) ? rsq*rsq*rsq : IE3;
}

// ---------- workspace float layout ----------
// [0]       dF : BNT*3      [BNT*3]   r  : BNT*3     [BNT*6]   gp : BNT*3
// [BNT*9]   mp : BNT*3      [BNT*12]  vp : BNT*3     [BNT*15]  gm : BNT
// [BNT*16]  mm : BNT        [BNT*17]  vm : BNT       [BNT*18]  bp : BNT*3
// [BNT*21]  bm : BNT        [BNT*22]  scalars: [0]=den [1]=loss [2]=best [3]=flag [4]=dF2

__global__ void k_scalars0(float* sc){
  sc[0]=0.f; sc[1]=0.f; sc[2]=__builtin_inff(); sc[3]=0.f; sc[4]=0.f;
}
__global__ void k_redpm(const float* __restrict__ pm, float* sc){
  int i = blockIdx.x*blockDim.x + threadIdx.x;
  if (i < BNT) atomicAdd(&sc[0], pm[i]);
}
__global__ void k_denfin(float* sc){ sc[0] = fmaxf(sc[0], 1.f); }
__global__ void k_zloss(float* sc){ sc[1] = 0.f; }
__global__ void k_best(float* sc){
  float l = sc[1]/sc[0];
  if (l < sc[2]){ sc[2]=l; sc[3]=1.f; } else sc[3]=0.f;
}

// ---------- delta_F via f32 WMMA ----------
// dF[i,c] = pm_i*G*m_i*(D_c - D_w*p_i,c),  W'[i,j] = m_j*miss_j/max(d,eps)^3 (diag 0)
__global__ __launch_bounds__(256) void k_deltaF(const float* __restrict__ cp,
      const float* __restrict__ cm, const float* __restrict__ pm,
      const float* __restrict__ Gp, float* __restrict__ dF, float* __restrict__ sc){
  __shared__ float jx[256], jy[256], jz[256], jw[256];   // jw = m_j*miss_j
  __shared__ float exs[2048];
  const int tid  = threadIdx.x;
  const int b    = blockIdx.x >> 3;
  const int ibase= (blockIdx.x & 7) * 128;
  const int wave = tid >> 5, lane = tid & 31;
  const int il   = lane & 15, kh = lane >> 4;
  const int i    = ibase + wave*16 + il;
  const float pix = cp[((b*NN)+i)*3+0];
  const float piy = cp[((b*NN)+i)*3+1];
  const float piz = cp[((b*NN)+i)*3+2];
  const float e0 = (il==0)?1.f:0.f, e1 = (il==1)?1.f:0.f;
  const float e2 = (il==2)?1.f:0.f, e3 = (il==3)?1.f:0.f;
  v8f acc = {0.f,0.f,0.f,0.f,0.f,0.f,0.f,0.f};

  for (int jt = 0; jt < NN; jt += 256){
    __syncthreads();
    int j = jt + tid;
    jx[tid] = cp[((b*NN)+j)*3+0];
    jy[tid] = cp[((b*NN)+j)*3+1];
    jz[tid] = cp[((b*NN)+j)*3+2];
    jw[tid] = cm[b*NN+j]*(1.0f - pm[b*NN+j]);
    __syncthreads();
    if (jt + 256 < NN) __builtin_prefetch(&cp[(size_t)((b*NN)+jt+256+tid)*3], 0, 0);
#pragma unroll 2
    for (int c0 = 0; c0 < 256; c0 += 4){
      float wv[2], bv[2];
#pragma unroll
      for (int q = 0; q < 2; q++){
        int jl = c0 + kh*2 + q;
        float px = jx[jl], py = jy[jl], pz = jz[jl];
        float dx = px-pix, dy = py-piy, dz = pz-piz;
        float r2 = dx*dx + dy*dy + dz*dz;
        float rsq = __builtin_amdgcn_rsqf(r2);
        float w = jw[jl]*inv3_of(r2, rsq);
        if (jt + jl == i) w = 0.f;                     // diagonal (select, no branch)
        wv[q] = w;
        bv[q] = e0*px + e1*py + e2*pz + e3;            // branchless B column build
      }
      v2f av = {wv[0], wv[1]};
      v2f bb = {bv[0], bv[1]};
      acc = __builtin_amdgcn_wmma_f32_16x16x4_f32(false, av, false, bb,
                                                  (short)0, acc, false, false);
    }
  }
#pragma unroll
  for (int rr = 0; rr < 8; rr++)
    exs[wave*256 + (rr + kh*8)*16 + il] = acc[rr];
  __syncthreads();
  if (tid < 128){
    int base = (tid >> 4)*256 + (tid & 15)*16;
    float Sx = exs[base+0], Sy = exs[base+1], Sz = exs[base+2], Ss = exs[base+3];
    int gi = b*NN + ibase + tid;
    float scale = pm[gi]*Gp[0]*cm[gi];
    float fx = scale*(Sx - Ss*cp[gi*3+0]);
    float fy = scale*(Sy - Ss*cp[gi*3+1]);
    float fz = scale*(Sz - Ss*cp[gi*3+2]);
    dF[gi*3+0]=fx; dF[gi*3+1]=fy; dF[gi*3+2]=fz;
    atomicAdd(&sc[4], fx*fx + fy*fy + fz*fz);
  }
}

// ---------- ghost init ----------
__global__ void k_init(const float* __restrict__ cp, const float* __restrict__ pm,
                       const float* __restrict__ gmi, float* __restrict__ ws){
  int idx = blockIdx.x*blockDim.x + threadIdx.x;
  if (idx >= BNT) return;
  float miss = 1.f - pm[idx];
  float g0 = gmi[0];
  float *gp = ws + (size_t)BNT*6,  *mp = ws + (size_t)BNT*9,  *vp = ws + (size_t)BNT*12;
  float *gm = ws + (size_t)BNT*15, *mm = ws + (size_t)BNT*16, *vm = ws + (size_t)BNT*17;
  float *bp = ws + (size_t)BNT*18, *bm = ws + (size_t)BNT*21;
#pragma unroll
  for (int c = 0; c < 3; c++){
    float z = nrm_f((unsigned)(idx*3 + c) + 0x9e3779b9u);
    float v = (cp[idx*3+c] + 0.1f*z)*miss;
    gp[idx*3+c]=v; bp[idx*3+c]=v; mp[idx*3+c]=0.f; vp[idx*3+c]=0.f;
  }
  gm[idx]=g0; bm[idx]=g0; mm[idx]=0.f; vm[idx]=0.f;
}

// ---------- forward via f32 WMMA: r = dF - gf, loss ----------
// gf[k,c] = G*mk*(D_c - D_w*pk_c), W[k,g] = s_g/max(d,eps)^3, B cols [gx,gy,gz,1]
__global__ __launch_bounds__(256) void k_fwd(const float* __restrict__ cp,
      const float* __restrict__ cm, const float* __restrict__ pm,
      const float* __restrict__ Gp, float* __restrict__ ws){
  __shared__ float gxl[256], gyl[256], gzl[256], gsl[256];
  __shared__ float exs[2048];
  const int tid  = threadIdx.x;
  const int b    = blockIdx.x >> 3;
  const int kbase= (blockIdx.x & 7) * 128;
  const int wave = tid >> 5, lane = tid & 31;
  const int il   = lane & 15, kh = lane >> 4;
  const int krow = kbase + wave*16 + il;
  const float* dF = ws;
  float* r        = ws + (size_t)BNT*3;
  const float* gp = ws + (size_t)BNT*6;
  const float* gm = ws + (size_t)BNT*15;
  float* sc       = ws + (size_t)BNT*22;
  const float pkx = cp[((b*NN)+krow)*3+0];
  const float pky = cp[((b*NN)+krow)*3+1];
  const float pkz = cp[((b*NN)+krow)*3+2];
  const float e0 = (il==0)?1.f:0.f, e1 = (il==1)?1.f:0.f;
  const float e2 = (il==2)?1.f:0.f, e3 = (il==3)?1.f:0.f;
  v8f acc = {0.f,0.f,0.f,0.f,0.f,0.f,0.f,0.f};

  for (int gt = 0; gt < NN; gt += 256){
    __syncthreads();
    int g = b*NN + gt + tid;
    gxl[tid]=gp[g*3]; gyl[tid]=gp[g*3+1]; gzl[tid]=gp[g*3+2];
    gsl[tid]=sp_f(gm[g])*(1.f - pm[g]);
    __syncthreads();
    if (gt + 256 < NN) __builtin_prefetch(&gp[(size_t)(b*NN + gt + 256 + tid)*3], 0, 0);
#pragma unroll 2
    for (int c0 = 0; c0 < 256; c0 += 4){
      float wv[2], bv[2];
#pragma unroll
      for (int q = 0; q < 2; q++){
        int jl = c0 + kh*2 + q;
        float gx = gxl[jl], gy = gyl[jl], gz = gzl[jl];
        float ux = gx-pkx, uy = gy-pky, uz = gz-pkz;
        float r2 = ux*ux + uy*uy + uz*uz;
        float rsq = __builtin_amdgcn_rsqf(r2);
        wv[q] = gsl[jl]*inv3_of(r2, rsq);
        bv[q] = e0*gx + e1*gy + e2*gz + e3;
      }
      v2f av = {wv[0], wv[1]};
      v2f bb = {bv[0], bv[1]};
      acc = __builtin_amdgcn_wmma_f32_16x16x4_f32(false, av, false, bb,
                                                  (short)0, acc, false, false);
    }
  }
#pragma unroll
  for (int rr = 0; rr < 8; rr++)
    exs[wave*256 + (rr + kh*8)*16 + il] = acc[rr];
  __syncthreads();
  if (tid < 128){
    int base = (tid >> 4)*256 + (tid & 15)*16;
    float Dx = exs[base+0], Dy = exs[base+1], Dz = exs[base+2], Dw = exs[base+3];
    int gi = b*NN + kbase + tid;
    float f = Gp[0]*cm[gi]*pm[gi];
    float rx = dF[gi*3+0] - f*(Dx - Dw*cp[gi*3+0]);
    float ry = dF[gi*3+1] - f*(Dy - Dw*cp[gi*3+1]);
    float rz = dF[gi*3+2] - f*(Dz - Dw*cp[gi*3+2]);
    r[gi*3+0]=rx; r[gi*3+1]=ry; r[gi*3+2]=rz;
    atomicAdd(&sc[1], rx*rx + ry*ry + rz*rz);
  }
}

// ---------- conditional best-snapshot ----------
__global__ void k_snap(float* __restrict__ ws){
  int idx = blockIdx.x*blockDim.x + threadIdx.x;
  const float* sc = ws + (size_t)BNT*22;
  if (idx >= BNT || sc[3] == 0.f) return;
  const float* gp = ws + (size_t)BNT*6;  const float* gm = ws + (size_t)BNT*15;
  float* bp = ws + (size_t)BNT*18;       float* bm = ws + (size_t)BNT*21;
  bp[idx*3]=gp[idx*3]; bp[idx*3+1]=gp[idx*3+1]; bp[idx*3+2]=gp[idx*3+2];
  bm[idx]=gm[idx];
}

// ---------- backward via dual f32 WMMA + fused Adam ----------
// W1[g,k]=base, B1 cols [rx,ry,rz,r.pk];  W2[g,k]=base*coef, B2 cols [pkx,pky,pkz,1]
// acc_c = D1_c - g_c*D2_w + D2_c ;  accm = g.D1_xyz - D1_w
__global__ __launch_bounds__(256) void k_bwd(const float* __restrict__ cp,
      const float* __restrict__ cm, const float* __restrict__ pm,
      const float* __restrict__ Gp, float* __restrict__ ws, float bc1, float bc2){
  __shared__ float kxl[256], kyl[256], kzl[256], kml[256];
  __shared__ float rxl[256], ryl[256], rzl[256], rpl[256];
  __shared__ float exs[4096];
  const int tid  = threadIdx.x;
  const int b    = blockIdx.x >> 3;
  const int gbase= (blockIdx.x & 7) * 128;
  const int wave = tid >> 5, lane = tid & 31;
  const int il   = lane & 15, kh = lane >> 4;
  const int grow = gbase + wave*16 + il;
  const float* rr_ = ws + (size_t)BNT*3;
  float* gpv = ws + (size_t)BNT*6;  float* mp  = ws + (size_t)BNT*9;  float* vp = ws + (size_t)BNT*12;
  float* gmv = ws + (size_t)BNT*15; float* mmv = ws + (size_t)BNT*16; float* vmv = ws + (size_t)BNT*17;
  const float* sc = ws + (size_t)BNT*22;
  const int gi0 = b*NN + grow;
  const float gx = gpv[gi0*3+0], gy = gpv[gi0*3+1], gz = gpv[gi0*3+2];
  const float e0 = (il==0)?1.f:0.f, e1 = (il==1)?1.f:0.f;
  const float e2 = (il==2)?1.f:0.f, e3 = (il==3)?1.f:0.f;
  v8f a1 = {0.f,0.f,0.f,0.f,0.f,0.f,0.f,0.f};
  v8f a2 = {0.f,0.f,0.f,0.f,0.f,0.f,0.f,0.f};

  for (int kt = 0; kt < NN; kt += 256){
    __syncthreads();
    int k = b*NN + kt + tid;
    float px = cp[k*3], py = cp[k*3+1], pz = cp[k*3+2];
    float rx = rr_[k*3], ry = rr_[k*3+1], rz = rr_[k*3+2];
    kxl[tid]=px; kyl[tid]=py; kzl[tid]=pz;
    kml[tid]=cm[k]*pm[k];
    rxl[tid]=rx; ryl[tid]=ry; rzl[tid]=rz;
    rpl[tid]=rx*px + ry*py + rz*pz;               // r . pk precomputed per column
    __syncthreads();
    if (kt + 256 < NN) __builtin_prefetch(&rr_[(size_t)(b*NN + kt + 256 + tid)*3], 0, 0);
    for (int c0 = 0; c0 < 256; c0 += 4){
      float w1[2], w2[2], b1[2], b2[2];
#pragma unroll
      for (int q = 0; q < 2; q++){
        int jl = c0 + kh*2 + q;
        float ux = gx-kxl[jl], uy = gy-kyl[jl], uz = gz-kzl[jl];
        float r2 = ux*ux + uy*uy + uz*uz;
        float rsq = __builtin_amdgcn_rsqf(r2);
        bool ok = (r2 >= EPS2);
        float inv3 = ok ? rsq*rsq*rsq : IE3;
        float base = kml[jl]*inv3;
        float ru = rxl[jl]*gx + ryl[jl]*gy + rzl[jl]*gz - rpl[jl];
        float coef = ok ? 3.f*ru*(rsq*rsq) : 0.f;  // clamp kills d-path grad
        w1[q] = base;
        w2[q] = base*coef;
        b1[q] = e0*rxl[jl] + e1*ryl[jl] + e2*rzl[jl] + e3*rpl[jl];
        b2[q] = e0*kxl[jl] + e1*kyl[jl] + e2*kzl[jl] + e3;
      }
      v2f av1 = {w1[0], w1[1]}, bb1 = {b1[0], b1[1]};
      v2f av2 = {w2[0], w2[1]}, bb2 = {b2[0], b2[1]};
      a1 = __builtin_amdgcn_wmma_f32_16x16x4_f32(false, av1, false, bb1,
                                                 (short)0, a1, false, false);
      a2 = __builtin_amdgcn_wmma_f32_16x16x4_f32(false, av2, false, bb2,
                                                 (short)0, a2, false, false);
    }
  }
#pragma unroll
  for (int rr2 = 0; rr2 < 8; rr2++){
    int off = wave*256 + (rr2 + kh*8)*16 + il;
    exs[off] = a1[rr2];
    exs[2048 + off] = a2[rr2];
  }
  __syncthreads();
  if (tid < 128){
    int base = (tid >> 4)*256 + (tid & 15)*16;
    float D1x = exs[base+0], D1y = exs[base+1], D1z = exs[base+2], D1w = exs[base+3];
    float D2x = exs[2048+base+0], D2y = exs[2048+base+1];
    float D2z = exs[2048+base+2], D2w = exs[2048+base+3];
    int idx = b*NN + gbase + tid;
    float gxr = gpv[idx*3+0], gyr = gpv[idx*3+1], gzr = gpv[idx*3+2];
    float gmval = gmv[idx];
    float miss = 1.f - pm[idx];
    float s = sp_f(gmval)*miss;
    float accx = D1x - gxr*D2w + D2x;
    float accy = D1y - gyr*D2w + D2y;
    float accz = D1z - gzr*D2w + D2z;
    float accm = gxr*D1x + gyr*D1y + gzr*D1z - D1w;
    float c2  = -2.f*Gp[0]/sc[0];
    float g1x = c2*s*accx, g1y = c2*s*accy, g1z = c2*s*accz;
    float g2  = c2*sig_f(gmval)*miss*accm;
    float m, v;
    m = B1C*mp[idx*3+0] + (1.f-B1C)*g1x; v = B2C*vp[idx*3+0] + (1.f-B2C)*g1x*g1x;
    mp[idx*3+0]=m; vp[idx*3+0]=v; gpv[idx*3+0] = gxr - LRC*(m/bc1)/(sqrtf(v/bc2)+AEPS);
    m = B1C*mp[idx*3+1] + (1.f-B1C)*g1y; v = B2C*vp[idx*3+1] + (1.f-B2C)*g1y*g1y;
    mp[idx*3+1]=m; vp[idx*3+1]=v; gpv[idx*3+1] = gyr - LRC*(m/bc1)/(sqrtf(v/bc2)+AEPS);
    m = B1C*mp[idx*3+2] + (1.f-B1C)*g1z; v = B2C*vp[idx*3+2] + (1.f-B2C)*g1z*g1z;
    mp[idx*3+2]=m; vp[idx*3+2]=v; gpv[idx*3+2] = gzr - LRC*(m/bc1)/(sqrtf(v/bc2)+AEPS);
    m = B1C*mmv[idx] + (1.f-B1C)*g2; v = B2C*vmv[idx] + (1.f-B2C)*g2*g2;
    mmv[idx]=m; vmv[idx]=v; gmv[idx] = gmval - LRC*(m/bc1)/(sqrtf(v/bc2)+AEPS);
  }
}

// ---------- compose outputs ----------
__global__ void k_final(const float* __restrict__ cp, const float* __restrict__ cm,
                        const float* __restrict__ pm, const float* __restrict__ ws,
                        float* __restrict__ out){
  int idx = blockIdx.x*blockDim.x + threadIdx.x;
  const float* bp = ws + (size_t)BNT*18;
  const float* bm = ws + (size_t)BNT*21;
  const float* sc = ws + (size_t)BNT*22;
  if (idx < BNT){
    float p = pm[idx], miss = 1.f - p;
    out[idx*3+0] = cp[idx*3+0]*p + bp[idx*3+0]*miss;
    out[idx*3+1] = cp[idx*3+1]*p + bp[idx*3+1]*miss;
    out[idx*3+2] = cp[idx*3+2]*p + bp[idx*3+2]*miss;
    out[(size_t)BNT*3 + idx] = cm[idx]*p + sp_f(bm[idx])*miss;
  }
  if (idx == 0) out[(size_t)BNT*4] = sc[4]/sc[0];
}

extern "C" void kernel_launch(void* const* d_in, const int* in_sizes, int n_in,
                              void* d_out, int out_size, void* d_ws, size_t ws_size,
                              hipStream_t stream) {
  (void)in_sizes; (void)n_in; (void)out_size; (void)ws_size;
  const float* cp  = (const float*)d_in[0];
  const float* cm  = (const float*)d_in[1];
  const float* pm  = (const float*)d_in[2];
  const float* Gp  = (const float*)d_in[3];
  const float* gmi = (const float*)d_in[4];
  float* ws  = (float*)d_ws;
  float* out = (float*)d_out;
  float* sc  = ws + (size_t)BNT*22;
  float* dF  = ws;

  k_scalars0<<<1, 1, 0, stream>>>(sc);
  k_redpm  <<<BNT/256, 256, 0, stream>>>(pm, sc);
  k_denfin <<<1, 1, 0, stream>>>(sc);
  k_deltaF <<<NB*8, 256, 0, stream>>>(cp, cm, pm, Gp, dF, sc);
  k_init   <<<BNT/256, 256, 0, stream>>>(cp, pm, gmi, ws);

  for (int t = 1; t <= 20; t++){
    k_zloss<<<1, 1, 0, stream>>>(sc);
    k_fwd  <<<NB*8, 256, 0, stream>>>(cp, cm, pm, Gp, ws);
    k_best <<<1, 1, 0, stream>>>(sc);
    k_snap <<<BNT/256, 256, 0, stream>>>(ws);
    float bc1 = 1.f - powf(B1C, (float)t);
    float bc2 = 1.f - powf(B2C, (float)t);
    k_bwd  <<<NB*8, 256, 0, stream>>>(cp, cm, pm, Gp, ws, bc1, bc2);
  }
  k_final<<<(BNT+255)/256, 256, 0, stream>>>(cp, cm, pm, ws, out);
}